// LIF_68788196213055
// MI455X (gfx1250) — compile-verified
//
#include <hip/hip_runtime.h>

// ---------------- problem constants (match reference) ----------------
constexpr int   Bk = 8192;
constexpr int   Lk = 4096;
constexpr float TAU_INV = 1.0f / 20.0f;
constexpr float THv     = 1.0f;
constexpr float K_SURR  = 10.0f;

// ---------------- tiling ----------------
constexpr int ROWS    = 32;          // rows per block == wave32
constexpr int T       = 64;          // timesteps per tile
constexpr int RSTRIDE = T + 4;       // 68 floats = 272B row stride: 16B aligned, low bank conflict
constexpr int NT      = Lk / T;      // 64 tiles
constexpr int ITER    = (ROWS * (T / 4)) / 32;  // 16 wave-wide float4 copy steps per tile
constexpr int NBUF    = 4;           // prefetch depth 3

#if __has_builtin(__builtin_amdgcn_global_load_async_to_lds_b128)
#define LIF_HAVE_ASYNC 1
#else
#define LIF_HAVE_ASYNC 0
#endif

// Vector-of-4-int pointee types matching the builtin's expected signature.
typedef int lif_v4i __attribute__((ext_vector_type(4)));
typedef __attribute__((address_space(1))) lif_v4i* lif_gptr;   // global (AS1)
typedef __attribute__((address_space(3))) lif_v4i* lif_lptr;   // LDS (AS3)

// s_wait_asynccnt needs an immediate; provide fixed thresholds.
__device__ __forceinline__ void lif_wait_async_32() {
#if LIF_HAVE_ASYNC
#if __has_builtin(__builtin_amdgcn_s_wait_asynccnt)
  __builtin_amdgcn_s_wait_asynccnt(32);
#else
  asm volatile("s_wait_asynccnt 0x20" ::: "memory");
#endif
#endif
}
__device__ __forceinline__ void lif_wait_async_16() {
#if LIF_HAVE_ASYNC
#if __has_builtin(__builtin_amdgcn_s_wait_asynccnt)
  __builtin_amdgcn_s_wait_asynccnt(16);
#else
  asm volatile("s_wait_asynccnt 0x10" ::: "memory");
#endif
#endif
}
__device__ __forceinline__ void lif_wait_async_0() {
#if LIF_HAVE_ASYNC
#if __has_builtin(__builtin_amdgcn_s_wait_asynccnt)
  __builtin_amdgcn_s_wait_asynccnt(0);
#else
  asm volatile("s_wait_asynccnt 0x0" ::: "memory");
#endif
#endif
}

__global__ __launch_bounds__(ROWS)
void LIF_scan_kernel(const float* __restrict__ I, float* __restrict__ out)
{
  __shared__ __align__(16) float buf[NBUF][ROWS * RSTRIDE];

  const int lane = threadIdx.x;                 // 0..31
  const int row0 = blockIdx.x * ROWS;

  float* spikes = out;
  float* hardl  = out + (size_t)Bk * Lk;
  float* softl  = hardl + Bk;

  // Issue one tile's global->LDS copy (wave-cooperative, fully coalesced:
  // each iteration moves 512 bytes, 256 contiguous from each of 2 consecutive rows).
  auto issue_tile = [&](int tile, int bsel) {
    const int t0 = tile * T;
#pragma unroll
    for (int i = 0; i < ITER; ++i) {
      const int c = i * 32 + lane;        // 0..511 float4 chunk id
      const int r = c >> 4;               // 16 chunks per row
      const int j = (c & 15) * 4;         // float index within row
      const float* g = I + (size_t)(row0 + r) * Lk + (t0 + j);
      float*       l = &buf[bsel][r * RSTRIDE + j];
#if LIF_HAVE_ASYNC
      __builtin_amdgcn_global_load_async_to_lds_b128(
          (lif_gptr)g, (lif_lptr)l, /*offset=*/0, /*cpol=*/0);
#else
      *(float4*)l = *(const float4*)g;    // global_load_b128 + ds_store_b128
#endif
    }
  };

  // Persistent per-row scan state (lane == row within block).
  float v      = 0.0f;
  float best   = -1.0f;
  int   bestIdx = 0;
  float sum_s  = 0.0f;
  float sum_st = 0.0f;
  float tf     = 0.0f;   // running float time index

  // Prologue: prefetch 3 tiles deep.
  issue_tile(0, 0);
  issue_tile(1, 1);
  issue_tile(2, 2);

  for (int k = 0; k < NT; ++k) {
    const int cur = k & (NBUF - 1);

    // Wait until tile k's async loads have landed (loads complete in order).
    const int pend_after = ((k + 3 < NT) ? (k + 3) : NT) - (k + 1);
    if (pend_after >= 2)      lif_wait_async_32();
    else if (pend_after == 1) lif_wait_async_16();
    else                      lif_wait_async_0();
    __syncthreads();   // single-wave WG: HW S_NOP, compiler memory fence

    // ---- sequential LIF scan over this tile (one row per lane) ----
    float* rb = &buf[cur][lane * RSTRIDE];
#pragma unroll 8
    for (int tt = 0; tt < T; ++tt) {
      const float Icur = rb[tt];
      v = v + (Icur - v * TAU_INV);                 // Euler step
      const float x = K_SURR * (v - THv);
      const float s = 1.0f / (1.0f + __expf(-x));   // surrogate spike
      v = v * (1.0f - s);                           // reset
      rb[tt] = s;                                   // stage spike in place
      sum_s  += s;
      sum_st  = fmaf(s, tf, sum_st);
      if (s > best) { best = s; bestIdx = k * T + tt; }
      tf += 1.0f;
    }
    __syncthreads();

    // ---- coalesced spike store from LDS staging ----
    const int t0 = k * T;
#pragma unroll
    for (int i = 0; i < ITER; ++i) {
      const int c = i * 32 + lane;
      const int r = c >> 4;
      const int j = (c & 15) * 4;
      const float4 sv = *(const float4*)&buf[cur][r * RSTRIDE + j];
      *(float4*)(spikes + (size_t)(row0 + r) * Lk + (t0 + j)) = sv;
    }

    // Prefetch tile k+3 into the buffer stored from in the previous iteration (k-1 slot).
    if (k + 3 < NT) issue_tile(k + 3, (k + 3) & (NBUF - 1));
  }

  // Per-row latency outputs (written as f32 per harness convention).
  hardl[row0 + lane] = (float)bestIdx;
  softl[row0 + lane] = sum_st / (sum_s + 1e-6f);
}

extern "C" void kernel_launch(void* const* d_in, const int* in_sizes, int n_in,
                              void* d_out, int out_size, void* d_ws, size_t ws_size,
                              hipStream_t stream) {
  (void)in_sizes; (void)n_in; (void)out_size; (void)d_ws; (void)ws_size;
  const float* I = (const float*)d_in[0];
  float* out = (float*)d_out;
  dim3 grid(Bk / ROWS);
  dim3 block(ROWS);
  hipLaunchKernelGGL(LIF_scan_kernel, grid, block, 0, stream, I, out);
}